// DotAttention_52243982188614
// MI455X (gfx1250) — compile-verified
//
#include <hip/hip_runtime.h>
#include <cstdint>

typedef float v2f __attribute__((ext_vector_type(2)));
typedef float v8f __attribute__((ext_vector_type(8)));

#define BATCH    64
#define TLEN     2048
#define FDIM     1024
#define NCHUNK   8
#define TCHUNK   (TLEN / NCHUNK)   // 256 timesteps per workgroup
#define TTILE    16                // timesteps per LDS tile (matches WMMA M)
#define NTILE    (TCHUNK / TTILE)  // 16 tiles per chunk
#define ROWW     (FDIM + 4)        // padded LDS row stride (words) -> conflict-free column reads
#define NTHREADS 256               // 8 waves (wave32)
#define KSLICE   (FDIM / 8)        // per-wave K range (128 features)

// ---- CDNA5 async global->LDS copy (ASYNCcnt-tracked), GVS addressing:
//      LDS byte address taken from a VGPR; global address = SGPR base + 32-bit VGPR offset
__device__ __forceinline__ void async_ld_b128(uint32_t lds_byte, uint32_t voff, const float* sbase) {
  asm volatile("global_load_async_to_lds_b128 %0, %1, %2"
               :: "v"(lds_byte), "v"(voff), "s"(sbase) : "memory");
}
__device__ __forceinline__ void wait_async_le16() {
  asm volatile("s_wait_asynccnt 16" ::: "memory");
}
__device__ __forceinline__ void wait_async_le0() {
  asm volatile("s_wait_asynccnt 0" ::: "memory");
}

// Pass 1: per (batch, t-chunk) workgroup. Streams encoder slice once through LDS
// (double-buffered async copies), computes scores with f32 WMMA, maintains online
// softmax (m, l) and a per-thread float4 context accumulator.
__global__ __launch_bounds__(NTHREADS) void attn_partial_kernel(
    const float* __restrict__ hd, const float* __restrict__ enc,
    float* __restrict__ ctx_part, float* __restrict__ ml_part)
{
  __shared__ __align__(16) float sE[2 * TTILE * ROWW];  // double-buffered tile
  __shared__ __align__(16) float sH[FDIM];              // decoder hidden (broadcast B operand)
  __shared__ float sPart[8 * TTILE];                    // per-wave partial scores
  __shared__ float sP[TTILE];                           // softmax weights for this tile
  __shared__ float sScale;                              // accumulator rescale
  __shared__ float sM, sL;                              // running max / running sum

  const int tid   = threadIdx.x;
  const int b     = blockIdx.y;
  const int chunk = blockIdx.x;
  const int t0    = chunk * TCHUNK;

  const float* gbase = enc + ((size_t)b * TLEN + t0) * FDIM;  // uniform -> SGPR pair

  // stage decoder hidden vector into LDS
  {
    float4 h4 = *(const float4*)(hd + (size_t)b * FDIM + 4 * tid);
    *(float4*)&sH[4 * tid] = h4;
  }
  if (tid == 0) { sM = -INFINITY; sL = 0.0f; }

  float4 acc = make_float4(0.f, 0.f, 0.f, 0.f);  // this thread owns features [4*tid, 4*tid+4)

  const uint32_t ldsE0 = (uint32_t)(uintptr_t)(&sE[0]);

  // prologue: issue async loads for tile 0 (16 rows x 1024 f32; thread tid copies 16B of each row)
  #pragma unroll
  for (int j = 0; j < TTILE; ++j) {
    uint32_t voff = (uint32_t)(((0 * TTILE + j) * FDIM + 4 * tid) * 4);
    uint32_t lds  = ldsE0 + (uint32_t)(j * ROWW * 4) + (uint32_t)(tid * 16);
    async_ld_b128(lds, voff, gbase);
  }

  const int wv = tid >> 5, lane = tid & 31;
  const int hi = lane >> 4, mrow = lane & 15;

  for (int it = 0; it < NTILE; ++it) {
    const int buf = it & 1;
    if (it + 1 < NTILE) {
      // issue next tile into the other buffer (its previous consumer finished last iteration)
      const uint32_t ldsN = ldsE0 + (uint32_t)(((it + 1) & 1) * TTILE * ROWW * 4);
      #pragma unroll
      for (int j = 0; j < TTILE; ++j) {
        uint32_t voff = (uint32_t)((((it + 1) * TTILE + j) * FDIM + 4 * tid) * 4);
        uint32_t lds  = ldsN + (uint32_t)(j * ROWW * 4) + (uint32_t)(tid * 16);
        async_ld_b128(lds, voff, gbase);
      }
      wait_async_le16();  // async loads complete in order -> tile 'it' landed
    } else {
      wait_async_le0();
    }
    __syncthreads();

    const float* Eb = &sE[buf * TTILE * ROWW];

    // ---- scores via V_WMMA_F32_16X16X4_F32: D = A(16t x 4f) * B(4f x 16, h broadcast) + C
    //      wave wv reduces its private K-range [wv*KSLICE, (wv+1)*KSLICE).
    //      The wave-dependent base is folded into the pointers so the loop trip count is a
    //      compile-time constant: uniform scalar loop, EXEC stays all-ones around the WMMAs.
    v8f c = {};
    const float* Er = Eb + mrow * ROWW + 2 * hi + wv * KSLICE;  // A: lane<16 -> K0,K1 ; lane>=16 -> K2,K3
    const float* Hr = sH + 2 * hi + wv * KSLICE;                // B: same K split, broadcast over N
    #pragma unroll 8
    for (int kk = 0; kk < KSLICE; kk += 4) {
      v2f a  = *(const v2f*)(Er + kk);
      v2f bb = *(const v2f*)(Hr + kk);
      c = __builtin_amdgcn_wmma_f32_16x16x4_f32(false, a, false, bb, (short)0, c, false, false);
    }
    // D layout: VGPR j holds M=j (lanes 0-15) / M=j+8 (lanes 16-31); all N columns equal.
    if (lane == 0) {
      #pragma unroll
      for (int j = 0; j < 8; ++j) sPart[wv * TTILE + j] = c[j];
    }
    if (lane == 16) {
      #pragma unroll
      for (int j = 0; j < 8; ++j) sPart[wv * TTILE + 8 + j] = c[j];
    }
    __syncthreads();

    // ---- online softmax update (wave 0; lanes duplicated across halves)
    if (wv == 0) {
      float s = 0.f;
      #pragma unroll
      for (int w = 0; w < 8; ++w) s += sPart[w * TTILE + mrow];
      float mx = s;
      #pragma unroll
      for (int off = 1; off < 16; off <<= 1) mx = fmaxf(mx, __shfl_xor(mx, off, 32));
      float mOld = sM;
      float mNew = fmaxf(mOld, mx);
      float p = __expf(s - mNew);
      float psum = p;
      #pragma unroll
      for (int off = 1; off < 16; off <<= 1) psum += __shfl_xor(psum, off, 32);
      if (lane < 16) sP[lane] = p;
      if (lane == 0) {
        float scl = __expf(mOld - mNew);
        sScale = scl;
        sM = mNew;
        sL = sL * scl + psum;
      }
    }
    __syncthreads();

    // ---- context accumulation: acc = acc*scale + sum_i p_i * E[t_i, my 4 features]
    const float scl = sScale;
    acc.x *= scl; acc.y *= scl; acc.z *= scl; acc.w *= scl;
    #pragma unroll
    for (int i = 0; i < TTILE; ++i) {
      float  pi = sP[i];
      float4 e  = *(const float4*)(Eb + i * ROWW + 4 * tid);
      acc.x = fmaf(pi, e.x, acc.x);
      acc.y = fmaf(pi, e.y, acc.y);
      acc.z = fmaf(pi, e.z, acc.z);
      acc.w = fmaf(pi, e.w, acc.w);
    }
    __syncthreads();  // all reads of buf done before it is refilled at it+1's issue
  }

  const size_t pidx = (size_t)b * NCHUNK + chunk;
  *(float4*)(ctx_part + pidx * FDIM + 4 * tid) = acc;
  if (tid == 0) {
    ml_part[pidx * 2 + 0] = sM;
    ml_part[pidx * 2 + 1] = sL;
  }
}

// Pass 2: combine the 8 per-chunk partials per batch with flash rescaling.
__global__ __launch_bounds__(NTHREADS) void attn_reduce_kernel(
    const float* __restrict__ ctx_part, const float* __restrict__ ml_part,
    float* __restrict__ out)
{
  const int b = blockIdx.x;
  const int tid = threadIdx.x;

  float m[NCHUNK];
  float M = -INFINITY;
  #pragma unroll
  for (int c = 0; c < NCHUNK; ++c) {
    m[c] = ml_part[((size_t)b * NCHUNK + c) * 2 + 0];
    M = fmaxf(M, m[c]);
  }
  float w[NCHUNK];
  float L = 0.f;
  #pragma unroll
  for (int c = 0; c < NCHUNK; ++c) {
    w[c] = __expf(m[c] - M);
    L += ml_part[((size_t)b * NCHUNK + c) * 2 + 1] * w[c];
  }
  const float inv = 1.0f / L;

  float4 o = make_float4(0.f, 0.f, 0.f, 0.f);
  #pragma unroll
  for (int c = 0; c < NCHUNK; ++c) {
    float4 x = *(const float4*)(ctx_part + ((size_t)b * NCHUNK + c) * FDIM + 4 * tid);
    o.x = fmaf(w[c], x.x, o.x);
    o.y = fmaf(w[c], x.y, o.y);
    o.z = fmaf(w[c], x.z, o.z);
    o.w = fmaf(w[c], x.w, o.w);
  }
  o.x *= inv; o.y *= inv; o.z *= inv; o.w *= inv;
  *(float4*)(out + (size_t)b * FDIM + 4 * tid) = o;
}

extern "C" void kernel_launch(void* const* d_in, const int* in_sizes, int n_in,
                              void* d_out, int out_size, void* d_ws, size_t ws_size,
                              hipStream_t stream) {
  (void)in_sizes; (void)n_in; (void)out_size; (void)ws_size;
  const float* hd  = (const float*)d_in[0];   // (64, 1024) f32
  const float* enc = (const float*)d_in[1];   // (64, 2048, 1024) f32
  float* out = (float*)d_out;                 // (64, 1024) f32

  float* ctx_part = (float*)d_ws;                                  // 64*8*1024 f32
  float* ml_part  = ctx_part + (size_t)BATCH * NCHUNK * FDIM;      // 64*8*2  f32

  dim3 g1(NCHUNK, BATCH);
  attn_partial_kernel<<<g1, dim3(NTHREADS), 0, stream>>>(hd, enc, ctx_part, ml_part);
  attn_reduce_kernel<<<dim3(BATCH), dim3(NTHREADS), 0, stream>>>(ctx_part, ml_part, out);
}